// PPO_26688926777562
// MI455X (gfx1250) — compile-verified
//
#include <hip/hip_runtime.h>
#include <stdint.h>

namespace {

constexpr int   kB       = 256;
constexpr int   kS       = 16384;
constexpr float kClip    = 0.2f;
constexpr float kGamma   = 1.0f;
constexpr float kDecay   = 0.95f;            // gamma * lam
constexpr int   kSeg     = 16;               // elements per thread per tile
constexpr int   kThreads = 256;
constexpr int   kTile    = kSeg * kThreads;  // 4096
constexpr int   kTiles   = kS / kTile;       // 4
constexpr int   kLStride = kSeg + 1;         // padded LDS stride -> conflict-free

constexpr double dpow(double b, int n) { double r = 1.0; for (int i = 0; i < n; ++i) r *= b; return r; }

// exact geometric-scan coefficients (compile-time doubles, rounded once)
constexpr float kA1    = (float)dpow(0.95, 16);   // segment-to-segment coeff
constexpr float kA1_2  = (float)dpow(0.95, 32);
constexpr float kA1_4  = (float)dpow(0.95, 64);
constexpr float kA1_8  = (float)dpow(0.95, 128);
constexpr float kA1_16 = (float)dpow(0.95, 256);
constexpr float kA1_32 = (float)dpow(0.95, 512); // wave-to-wave coeff

constexpr float kLog2A1 = -1.18400930310043087f;  // 16 * log2(0.95)

__device__ __forceinline__ float a1pow(int k) {   // (0.95^16)^k, k >= 0 (underflows to 0 cleanly)
  return exp2f((float)k * kLog2A1);
}

} // namespace

extern "C" __global__ __launch_bounds__(256)
void ppo_fused(const float* __restrict__ rew, const float* __restrict__ val,
               const float* __restrict__ nlp, const float* __restrict__ olp,
               const float* __restrict__ nv,  const int* __restrict__ msk,
               float* __restrict__ out)
{
  __shared__ float LR[kThreads * kLStride];   // rewards tile, padded
  __shared__ float LV[kThreads * kLStride];   // values  tile, padded
  __shared__ float wsum[8];                   // per-wave zero-carry scan heads
  __shared__ float kslot;                     // cross-tile carry
  __shared__ float red[8][4];                 // final reduction scratch

  const int tid  = threadIdx.x;
  const int lane = tid & 31;
  const int wave = tid >> 5;
  const int row  = blockIdx.x;
  const int rowBase = row * kS;

  float K    = 0.f;                 // gae at right boundary of current tile (bootstrap 0)
  float accP = 0.f, accV = 0.f, accM = 0.f;

  for (int T = kTiles - 1; T >= 0; --T) {
    const int tileBase = rowBase + T * kTile;

    // ---- stage rewards/values into padded LDS via CDNA5 async global->LDS copies ----
    #pragma unroll
    for (int i = 0; i < kSeg; ++i) {
      const int g  = tid + i * kThreads;              // coalesced across lanes
      const int li = kLStride * (g >> 4) + (g & 15);  // padded slot for owning thread
      const uint32_t lR = (uint32_t)(uintptr_t)&LR[li];  // low 32 bits == LDS address (aperture rule)
      const uint32_t lV = (uint32_t)(uintptr_t)&LV[li];
      const float* aR = rew + tileBase + g;
      const float* aV = val + tileBase + g;
      asm volatile("global_load_async_to_lds_b32 %0, %1, off" :: "v"(lR), "v"(aR) : "memory");
      asm volatile("global_load_async_to_lds_b32 %0, %1, off" :: "v"(lV), "v"(aV) : "memory");
    }
    asm volatile("s_wait_asynccnt 0" ::: "memory");
    __syncthreads();

    // value just past this thread's segment (bootstrap V_S = 0)
    float vend;
    if (tid < kThreads - 1) vend = LV[kLStride * (tid + 1)];
    else                    vend = (T < kTiles - 1) ? val[tileBase + kTile] : 0.f;

    // ---- pass A: local reverse scan with zero carry -> s = gae-local at segment start ----
    float s = 0.f;
    {
      float vn = vend;
      #pragma unroll
      for (int j = kSeg - 1; j >= 0; --j) {
        const float v = LV[kLStride * tid + j];
        const float r = LR[kLStride * tid + j];
        const float delta = r + kGamma * vn - v;
        s  = delta + kDecay * s;
        vn = v;
      }
    }

    // ---- wave-level weighted suffix scan: y_l = sum_{m>=l} A1^(m-l) * s_m ----
    float y = s, u;
    u = __shfl(y, lane + 1);  if (lane < 31) y += kA1    * u;
    u = __shfl(y, lane + 2);  if (lane < 30) y += kA1_2  * u;
    u = __shfl(y, lane + 4);  if (lane < 28) y += kA1_4  * u;
    u = __shfl(y, lane + 8);  if (lane < 24) y += kA1_8  * u;
    u = __shfl(y, lane + 16); if (lane < 16) y += kA1_16 * u;

    if (lane == 0) wsum[wave] = y;
    __syncthreads();

    // ---- cross-wave combine: Hn = zero-tile-carry gae at start of wave+1 ----
    float Hn = 0.f;
    for (int w = 7; w > wave; --w) Hn = wsum[w] + kA1_32 * Hn;

    // carry into this thread's segment = gae at its right boundary
    const float yn = __shfl(y, lane + 1);
    const float Cz = (lane < 31) ? (yn + a1pow(31 - lane) * Hn) : Hn;
    const float C  = Cz + a1pow(255 - tid) * K;     // add inter-tile carry (exact powers)

    if (tid == 0) kslot = y + kA1_32 * Hn;          // gae at tile start -> next tile's K

    // ---- pass B: recompute local scan, apply carry, fused PPO math ----
    {
      const int segBase = tileBase + tid * kSeg;
      float s2 = 0.f, vn = vend, cp = 1.f;
      #pragma unroll
      for (int jj = kSeg - 4; jj >= 0; jj -= 4) {
        const float4 a4 = *(const float4*)(nlp + segBase + jj);
        const float4 b4 = *(const float4*)(olp + segBase + jj);
        const float4 w4 = *(const float4*)(nv  + segBase + jj);
        const int4   m4 = *(const int4*)  (msk + segBase + jj);
        const float av[4] = {a4.x, a4.y, a4.z, a4.w};
        const float bv[4] = {b4.x, b4.y, b4.z, b4.w};
        const float wv[4] = {w4.x, w4.y, w4.z, w4.w};
        const int   mv[4] = {m4.x, m4.y, m4.z, m4.w};
        #pragma unroll
        for (int e = 3; e >= 0; --e) {
          const int j = jj + e;
          const float v = LV[kLStride * tid + j];
          const float r = LR[kLStride * tid + j];
          const float delta = r + kGamma * vn - v;
          s2 = delta + kDecay * s2;
          vn = v;
          cp *= kDecay;                              // cp = 0.95^(16-j), exact powers
          const float adv = s2 + cp * C;
          const float ret = adv + v;
          const float ratio = expf(av[e] - bv[e]);
          const float rcl   = fminf(fmaxf(ratio, 1.f - kClip), 1.f + kClip);
          const float pl    = -fminf(ratio * adv, rcl * adv);
          const float m     = (float)mv[e];
          accP += pl * m;
          const float d2 = wv[e] - ret;
          accV += d2 * d2 * m;
          accM += m;
        }
      }
    }
    __syncthreads();     // pass-B LDS reads done; kslot published
    K = kslot;
  }

  // ---- deterministic block reduction of masked sums ----
  #pragma unroll
  for (int d = 16; d >= 1; d >>= 1) {
    accP += __shfl_xor(accP, d);
    accV += __shfl_xor(accV, d);
    accM += __shfl_xor(accM, d);
  }
  if (lane == 0) { red[wave][0] = accP; red[wave][1] = accV; red[wave][2] = accM; }
  __syncthreads();
  if (tid == 0) {
    float P = 0.f, V = 0.f, M = 0.f;
    #pragma unroll
    for (int w = 0; w < 8; ++w) { P += red[w][0]; V += red[w][1]; M += red[w][2]; }
    const float den = fmaxf(M, 1e-8f);
    out[row]      = P / den;   // p_loss
    out[kB + row] = V / den;   // v_loss
  }
}

extern "C" void kernel_launch(void* const* d_in, const int* in_sizes, int n_in,
                              void* d_out, int out_size, void* d_ws, size_t ws_size,
                              hipStream_t stream) {
  (void)in_sizes; (void)n_in; (void)out_size; (void)d_ws; (void)ws_size;
  const float* rew = (const float*)d_in[0];
  const float* val = (const float*)d_in[1];
  const float* nlp = (const float*)d_in[2];
  const float* olp = (const float*)d_in[3];
  const float* nv  = (const float*)d_in[4];
  const int*   msk = (const int*)d_in[5];
  float* out = (float*)d_out;
  hipLaunchKernelGGL(ppo_fused, dim3(kB), dim3(kThreads), 0, stream,
                     rew, val, nlp, olp, nv, msk, out);
}